// GCN_49323404427479
// MI455X (gfx1250) — compile-verified
//
#include <hip/hip_runtime.h>
#include <hip/hip_bf16.h>

#define N_NODES 100000
#define DIM     256
#define N_EDGES 800000
#define L2EPS   1e-12f

typedef __attribute__((ext_vector_type(2))) float v2f;
typedef __attribute__((ext_vector_type(8))) float v8f;

// ---------------------------------------------------------------------------
// Kernel 1: zero the aggregation buffer (d_out) and form vals = adj + adj_w
// ---------------------------------------------------------------------------
__global__ __launch_bounds__(256)
void gcn_init_kernel(const float* __restrict__ adj, const float* __restrict__ adjw,
                     float* __restrict__ vals, float* __restrict__ out,
                     int nd, int e) {
    int idx = blockIdx.x * blockDim.x + threadIdx.x;
    if (idx < nd) out[idx] = 0.0f;
    if (idx < e)  vals[idx] = adj[idx] + adjw[idx];
}

// ---------------------------------------------------------------------------
// Kernel 2: support = x @ W via V_WMMA_F32_16X16X4_F32, one 16x16 tile / wave
// A-frag layout (16x4 f32): lanes 0-15 hold rows M=0..15; v0 = K={0|2},
// v1 = K={1|3} selected by lane half.  B-frag (4x16): v0 = row K={0|2},
// v1 = row K={1|3}, N = lane&15.  C/D: VGPR i -> M = i + 8*(lane>=16), N=lane&15.
// ---------------------------------------------------------------------------
__global__ __launch_bounds__(128)
void gcn_gemm_wmma_kernel(const float* __restrict__ x, const float* __restrict__ w,
                          float* __restrict__ support) {
    const int wave   = blockIdx.x * (blockDim.x >> 5) + (threadIdx.x >> 5);
    const int lane   = threadIdx.x & 31;
    const int laneHi = lane >> 4;    // 0 or 1 (lane half)
    const int laneLo = lane & 15;

    const int nTile = wave & 15;     // 256/16 = 16 column tiles
    const int mTile = wave >> 4;     // row tile (wave-uniform -> EXEC all 1s)
    if (mTile >= N_NODES / 16) return;

    const int m0 = mTile * 16;
    const int n0 = nTile * 16;

    // Per-lane base pointers.
    const float* aBase = x + (size_t)(m0 + laneLo) * DIM + 2 * laneHi; // row m, k-offset
    const float* bBase = w + (size_t)(2 * laneHi) * DIM + n0 + laneLo; // row k, col n

    v8f acc0 = {};
    v8f acc1 = {};

    #pragma unroll 4
    for (int kk = 0; kk < DIM; kk += 8) {
        // chain 0: K = kk .. kk+3
        v2f a0 = *(const v2f*)(aBase + kk);                    // contiguous -> b64 load
        v2f b0;
        b0.x = bBase[(size_t)(kk + 0) * DIM];
        b0.y = bBase[(size_t)(kk + 1) * DIM];
        acc0 = __builtin_amdgcn_wmma_f32_16x16x4_f32(
            false, a0, false, b0, (short)0, acc0, false, false);

        // chain 1: K = kk+4 .. kk+7 (independent accumulator hides WMMA RAW NOPs)
        v2f a1 = *(const v2f*)(aBase + kk + 4);
        v2f b1;
        b1.x = bBase[(size_t)(kk + 4) * DIM];
        b1.y = bBase[(size_t)(kk + 5) * DIM];
        acc1 = __builtin_amdgcn_wmma_f32_16x16x4_f32(
            false, a1, false, b1, (short)0, acc1, false, false);
    }

    v8f sum = acc0 + acc1;

    float* oBase = support + (size_t)(m0 + 8 * laneHi) * DIM + n0 + laneLo;
    #pragma unroll
    for (int i = 0; i < 8; ++i)
        oBase[(size_t)i * DIM] = sum[i];
}

// ---------------------------------------------------------------------------
// Kernel 3: SpMM scatter — one wave per edge, lanes sweep the 256-dim row.
// out[dst, :] += vals[e] * support[src, :]   (native f32 atomics at L2)
// ---------------------------------------------------------------------------
__global__ __launch_bounds__(256)
void gcn_spmm_kernel(const int* __restrict__ ei, const float* __restrict__ vals,
                     const float* __restrict__ support, float* __restrict__ out) {
    const int e    = blockIdx.x * 8 + (threadIdx.x >> 5);
    const int lane = threadIdx.x & 31;
    if (e >= N_EDGES) return;

    const int dst = ei[e];             // edge_index[0, e]
    const int src = ei[N_EDGES + e];   // edge_index[1, e]
    const float v = vals[e];

    const float* sRow = support + (size_t)src * DIM;
    float*       oRow = out     + (size_t)dst * DIM;

    #pragma unroll
    for (int d = lane; d < DIM; d += 32) {
        unsafeAtomicAdd(&oRow[d], v * sRow[d]);
    }
}

// ---------------------------------------------------------------------------
// Kernel 4: in-place row L2 normalization, one 256-thread block per row.
// ---------------------------------------------------------------------------
__global__ __launch_bounds__(256)
void gcn_normalize_kernel(float* __restrict__ out) {
    __shared__ float red[8];
    const int row = blockIdx.x;
    const int tid = threadIdx.x;

    float v = out[(size_t)row * DIM + tid];
    float s = v * v;

    #pragma unroll
    for (int off = 16; off > 0; off >>= 1)
        s += __shfl_xor(s, off, 32);
    if ((tid & 31) == 0) red[tid >> 5] = s;
    __syncthreads();

    float total = 0.0f;
    #pragma unroll
    for (int i = 0; i < 8; ++i) total += red[i];

    const float scale = 1.0f / fmaxf(sqrtf(total), L2EPS);
    out[(size_t)row * DIM + tid] = v * scale;
}

// ---------------------------------------------------------------------------
extern "C" void kernel_launch(void* const* d_in, const int* in_sizes, int n_in,
                              void* d_out, int out_size, void* d_ws, size_t ws_size,
                              hipStream_t stream) {
    const float* x    = (const float*)d_in[0];  // [N, D]
    const float* w    = (const float*)d_in[1];  // [D, D]
    const float* adj  = (const float*)d_in[2];  // [E]
    const float* adjw = (const float*)d_in[3];  // [E]
    const int*   ei   = (const int*)  d_in[4];  // [2, E]

    float* out     = (float*)d_out;                         // [N, D] agg -> normalized
    float* support = (float*)d_ws;                          // [N, D]
    float* vals    = support + (size_t)N_NODES * DIM;       // [E]

    const int nd = N_NODES * DIM;

    // 1) zero accumulator + fuse edge weights
    gcn_init_kernel<<<(nd + 255) / 256, 256, 0, stream>>>(adj, adjw, vals, out,
                                                          nd, N_EDGES);

    // 2) support = x @ W   (100000 16x16 wave-tiles, 4 waves per block)
    const int tiles = (N_NODES / 16) * (DIM / 16);          // 100000
    gcn_gemm_wmma_kernel<<<(tiles + 3) / 4, 128, 0, stream>>>(x, w, support);

    // 3) scatter-add messages (one wave per edge)
    gcn_spmm_kernel<<<(N_EDGES + 7) / 8, 256, 0, stream>>>(ei, vals, support, out);

    // 4) row-wise L2 normalize in place
    gcn_normalize_kernel<<<N_NODES, 256, 0, stream>>>(out);
}